// MiaoBlock_6734508720702
// MI455X (gfx1250) — compile-verified
//
#include <hip/hip_runtime.h>
#include <math.h>

#define N_ATOMS 20000
#define N_EDGES 200000
#define CCH 64
#define NB 8
#define NK 11
#define NSLAB 13

typedef float v2f __attribute__((ext_vector_type(2)));
typedef float v8f __attribute__((ext_vector_type(8)));

__device__ __forceinline__ float silu_f(float x) { return x / (1.0f + __expf(-x)); }

// ---------------------------------------------------------------------------
// Edge kernel: one wave (32 lanes) per edge, 2 channels per lane.
// Computes radial filters fc[11][c], contracts gathered h0/h1/h2 with r-hat
// tensors, atomically scatters message slabs into AW[n][13][64].
// ---------------------------------------------------------------------------
__global__ void edge_kernel(const float* __restrict__ node0,
                            const float* __restrict__ node1,
                            const float* __restrict__ node2,
                            const float* __restrict__ rij,
                            const int*   __restrict__ idx_i,
                            const int*   __restrict__ idx_j,
                            const float* __restrict__ rbf_w,
                            float* __restrict__ AW) {
  __shared__ float wlds[NK * NB * CCH];   // 22.5 KB of rbf mixing weights
  const int tid = threadIdx.x;
  for (int i = tid; i < NK * NB * CCH; i += 256) wlds[i] = rbf_w[i];
  __syncthreads();

  const int wave = tid >> 5, lane = tid & 31;
  const int e = blockIdx.x * 8 + wave;
  if (e >= N_EDGES) return;

  const float rx = rij[e * 3 + 0], ry = rij[e * 3 + 1], rz = rij[e * 3 + 2];
  const float d = sqrtf(rx * rx + ry * ry + rz * rz);
  const float inv = 1.0f / (d + 1e-12f);
  const float u0 = rx * inv, u1 = ry * inv, u2 = rz * inv;
  const float u[3] = {u0, u1, u2};
  const float dc = fminf(d, 5.0f);
  const float fcut = 0.5f * (__cosf(3.14159265358979323846f * dc * 0.2f) + 1.0f);

  float rbf[NB];
#pragma unroll
  for (int b = 0; b < NB; ++b) {
    const float mu = 5.0f * (float)b / 7.0f;
    const float t = d - mu;
    rbf[b] = __expf(-2.0f * t * t) * fcut;
  }

  const int ii = idx_i[e], jj = idx_j[e];

#pragma unroll
  for (int ci = 0; ci < 2; ++ci) {
    const int c = lane + 32 * ci;
    float fc[NK];
#pragma unroll
    for (int k = 0; k < NK; ++k) {
      float acc = 0.0f;
#pragma unroll
      for (int b = 0; b < NB; ++b)
        acc = fmaf(rbf[b], wlds[(k * NB + b) * CCH + c], acc);
      fc[k] = acc;
    }

    const float h0v = node0[(size_t)jj * CCH + c];
    float h1v[3], h2v[9];
#pragma unroll
    for (int x = 0; x < 3; ++x) h1v[x] = node1[((size_t)jj * CCH + c) * 3 + x];
#pragma unroll
    for (int x = 0; x < 9; ++x) h2v[x] = node2[((size_t)jj * CCH + c) * 9 + x];

    const float s1 = h1v[0] * u0 + h1v[1] * u1 + h1v[2] * u2;
    float v2[3];
#pragma unroll
    for (int x = 0; x < 3; ++x)
      v2[x] = h2v[3 * x + 0] * u0 + h2v[3 * x + 1] * u1 + h2v[3 * x + 2] * u2;
    const float quad = v2[0] * u0 + v2[1] * u1 + v2[2] * u2;

    float* dst = AW + (size_t)ii * NSLAB * CCH + c;

    // way-0 message
    atomicAdd(dst + 0 * CCH, fc[0] * h0v + fc[4] * s1 + fc[9] * quad);
    // way-1 messages
#pragma unroll
    for (int x = 0; x < 3; ++x) {
      const float m1 = fc[1] * h0v * u[x] + fc[3] * h1v[x] + fc[6] * s1 * u[x] + fc[8] * v2[x];
      atomicAdd(dst + (1 + x) * CCH, m1);
    }
    // way-2 messages: (fc2*h0*u[x] + fc5*h1[x] + fc10*v2[x])*u[y] + fc7*h2[x][y]
#pragma unroll
    for (int x = 0; x < 3; ++x) {
      const float pre = fc[2] * h0v * u[x] + fc[5] * h1v[x] + fc[10] * v2[x];
#pragma unroll
      for (int y = 0; y < 3; ++y)
        atomicAdd(dst + (4 + 3 * x + y) * CCH, pre * u[y] + fc[7] * h2v[3 * x + y]);
    }
  }
}

// ---------------------------------------------------------------------------
// Atom kernel: 16-atom tile per block, 8 waves. WMMA f32 16x16x4 GEMM for the
// per-way 64x64 self-interaction, then gated-SiLU nonlinearity + residual.
// ---------------------------------------------------------------------------
__global__ void atom_kernel(const float* __restrict__ AW,
                            const float* __restrict__ node0,
                            const float* __restrict__ node1,
                            const float* __restrict__ node2,
                            const float* __restrict__ w0,
                            const float* __restrict__ b0,
                            const float* __restrict__ w1,
                            const float* __restrict__ w2,
                            const float* __restrict__ nlw1,
                            const float* __restrict__ nlb1,
                            const float* __restrict__ nlw2,
                            const float* __restrict__ nlb2,
                            float* __restrict__ out0,
                            float* __restrict__ out1,
                            float* __restrict__ out2) {
  __shared__ float A[16 * NSLAB * CCH];   // 52 KB: staged accumulators (/10)
  __shared__ float S[16 * NSLAB * CCH];   // 52 KB: GEMM results
  const int tid = threadIdx.x;
  const int n0 = blockIdx.x * 16;

  // Stage the fully contiguous [16 atoms][13 slabs][64 ch] tile, scaled by 1/NORM_FACTOR.
  {
    const float4* src = (const float4*)(AW + (size_t)n0 * NSLAB * CCH);
    float4* dstA = (float4*)A;
    for (int i = tid; i < 16 * NSLAB * CCH / 4; i += 256) {
      float4 v = src[i];
      v.x *= 0.1f; v.y *= 0.1f; v.z *= 0.1f; v.w *= 0.1f;
      dstA[i] = v;
    }
  }
  __syncthreads();

  const int wave = tid >> 5, lane = tid & 31;
  const int row = lane & 15, hi = lane >> 4;

  // 13 slabs x 4 channel-tiles = 52 WMMA tasks over 8 waves.
  for (int t = wave; t < NSLAB * 4; t += 8) {
    const int slab = t >> 2, ntile = t & 3;
    const float* W = (slab == 0) ? w0 : ((slab < 4) ? w1 : w2);
    const int ncol = ntile * 16 + row;
    const float* Arow = A + (row * NSLAB + slab) * CCH;

    v8f acc = {0.f, 0.f, 0.f, 0.f, 0.f, 0.f, 0.f, 0.f};
#pragma unroll
    for (int kk = 0; kk < 16; ++kk) {
      const int k0 = kk * 4 + 2 * hi;              // lanes 0-15: K={4k,4k+1}; 16-31: K={4k+2,4k+3}
      const float2 af = *(const float2*)(Arow + k0);
      v2f a; a.x = af.x; a.y = af.y;
      v2f b; b.x = W[k0 * CCH + ncol]; b.y = W[(k0 + 1) * CCH + ncol];
      acc = __builtin_amdgcn_wmma_f32_16x16x4_f32(
          /*neg_a=*/false, a, /*neg_b=*/false, b,
          /*c_mod=*/(short)0, acc, /*reuse_a=*/false, /*reuse_b=*/false);
    }
    // D layout: VGPR v holds (M = v + 8*hi, N = lane%16)
#pragma unroll
    for (int v = 0; v < 8; ++v) {
      const int m = v + 8 * hi;
      S[(m * NSLAB + slab) * CCH + ntile * 16 + row] = acc[v];
    }
  }
  __syncthreads();

  // Nonlinearity + residual + store: 1024 (atom, channel) pairs.
  for (int p = tid; p < 16 * CCH; p += 256) {
    const int a = p >> 6, c = p & 63;
    const int n = n0 + a;
    const float* Sr = S + (a * NSLAB) * CCH + c;

    const float s0 = Sr[0] + b0[c];
    out0[(size_t)n * CCH + c] = node0[(size_t)n * CCH + c] + silu_f(s0);

    float s1v[3]; float nn1 = 1e-6f;
#pragma unroll
    for (int x = 0; x < 3; ++x) { s1v[x] = Sr[(1 + x) * CCH]; nn1 += s1v[x] * s1v[x]; }
    const float g1 = silu_f(sqrtf(nn1) * nlw1[c] + nlb1[c]);
#pragma unroll
    for (int x = 0; x < 3; ++x) {
      const size_t o = ((size_t)n * CCH + c) * 3 + x;
      out1[o] = node1[o] + s1v[x] * g1;
    }

    float s2v[9]; float nn2 = 1e-6f;
#pragma unroll
    for (int x = 0; x < 9; ++x) { s2v[x] = Sr[(4 + x) * CCH]; nn2 += s2v[x] * s2v[x]; }
    const float g2 = silu_f(sqrtf(nn2) * nlw2[c] + nlb2[c]);
#pragma unroll
    for (int x = 0; x < 9; ++x) {
      const size_t o = ((size_t)n * CCH + c) * 9 + x;
      out2[o] = node2[o] + s2v[x] * g2;
    }
  }
}

// ---------------------------------------------------------------------------
extern "C" void kernel_launch(void* const* d_in, const int* in_sizes, int n_in,
                              void* d_out, int out_size, void* d_ws, size_t ws_size,
                              hipStream_t stream) {
  const float* node0 = (const float*)d_in[0];
  const float* node1 = (const float*)d_in[1];
  const float* node2 = (const float*)d_in[2];
  const float* edge0 = (const float*)d_in[3];
  const float* rij   = (const float*)d_in[4];
  const int*   idx_i = (const int*)d_in[5];
  const int*   idx_j = (const int*)d_in[6];
  const float* rbf_w = (const float*)d_in[7];
  const float* si_w0 = (const float*)d_in[8];
  const float* si_b0 = (const float*)d_in[9];
  const float* si_w1 = (const float*)d_in[10];
  const float* si_w2 = (const float*)d_in[11];
  const float* nl_w1 = (const float*)d_in[12];
  const float* nl_b1 = (const float*)d_in[13];
  const float* nl_w2 = (const float*)d_in[14];
  const float* nl_b2 = (const float*)d_in[15];

  float* out0 = (float*)d_out;                               // [N, C]
  float* out1 = out0 + (size_t)N_ATOMS * CCH;                // [N, C, 3]
  float* out2 = out1 + (size_t)N_ATOMS * CCH * 3;            // [N, C, 3, 3]
  float* out3 = out2 + (size_t)N_ATOMS * CCH * 9;            // [E, C] passthrough

  float* AW = (float*)d_ws;                                  // [N][13][64]
  const size_t aw_bytes = (size_t)N_ATOMS * NSLAB * CCH * sizeof(float);
  hipMemsetAsync(AW, 0, aw_bytes, stream);

  edge_kernel<<<N_EDGES / 8, 256, 0, stream>>>(node0, node1, node2, rij,
                                               idx_i, idx_j, rbf_w, AW);

  atom_kernel<<<N_ATOMS / 16, 256, 0, stream>>>(AW, node0, node1, node2,
                                                si_w0, si_b0, si_w1, si_w2,
                                                nl_w1, nl_b1, nl_w2, nl_b2,
                                                out0, out1, out2);

  hipMemcpyAsync(out3, edge0, (size_t)N_EDGES * CCH * sizeof(float),
                 hipMemcpyDeviceToDevice, stream);
}